// ISABPEERMetaNet_18356690223760
// MI455X (gfx1250) — compile-verified
//
#include <hip/hip_runtime.h>

#define N_ELEM 2097152
#define GRID   2048
#define BLOCK  256
#define NWAVE  8
#define SCALE_D 0.35355339059327373f   // 1/sqrt(8)
#define RESCALE 0.1f

// workspace layout (float offsets)
#define WS_S1  0      // 16*8 softmax numerators
#define WS_S0  128    // 16   softmax denominators
#define WS_IQ  144    // 16*8 inducing queries
#define WS_WRP 272    // 8*2  read_q_W @ input_proj_W
#define WS_BRP 288    // 8    read_q_W @ input_proj_b

typedef float v2f __attribute__((ext_vector_type(2)));
typedef float v8f __attribute__((ext_vector_type(8)));

// D = A(16x4 f32) * B(4x16 f32) + C(16x16 f32), full fp32 precision.
static __device__ __forceinline__ v8f wmma4(v2f a, v2f b, v8f c) {
  return __builtin_amdgcn_wmma_f32_16x16x4_f32(false, a, false, b, (short)0, c, false, false);
}

// Intra-wave LDS producer->consumer ordering: LDS ops are in-order per wave,
// s_wait_dscnt 0 guarantees completion; "memory" clobber stops compiler motion.
static __device__ __forceinline__ void wave_lds_fence() {
  asm volatile("s_wait_dscnt 0" ::: "memory");
}

// ---------------------------------------------------------------------------
// Setup: iq = inducing_points @ induce_q_W^T ; composite read-query matrix
// Wrp = read_q_W @ input_proj_W, brp = read_q_W @ input_proj_b ; zero S sums.
// ---------------------------------------------------------------------------
__global__ void k_setup(const float* __restrict__ ip, const float* __restrict__ Wq,
                        const float* __restrict__ Wr, const float* __restrict__ Wp,
                        const float* __restrict__ bp, float* __restrict__ ws) {
  int t = threadIdx.x;
  if (t < 144) ws[WS_S1 + t] = 0.0f;               // S1 (128) + S0 (16)
  if (t < 128) {
    int m = t >> 3, d = t & 7;
    float acc = 0.0f;
    #pragma unroll
    for (int k = 0; k < 8; ++k) acc += ip[m * 8 + k] * Wq[d * 8 + k];
    ws[WS_IQ + t] = acc;
  }
  if (t < 16) {
    int j = t >> 1, i = t & 1;
    float acc = 0.0f;
    #pragma unroll
    for (int k = 0; k < 8; ++k) acc += Wr[j * 8 + k] * Wp[k * 2 + i];
    ws[WS_WRP + t] = acc;
  }
  if (t >= 16 && t < 24) {
    int j = t - 16;
    float acc = 0.0f;
    #pragma unroll
    for (int k = 0; k < 8; ++k) acc += Wr[j * 8 + k] * bp[k];
    ws[WS_BRP + j] = acc;
  }
}

// ---------------------------------------------------------------------------
// Pass 1: global softmax accumulators.
// Per wave iteration (32 elements = two 16-row tiles):
//   x (per-lane) -> [ik|iv] = x*[Wk^T|Wv^T] (WMMA) -> E = iq*ik^T (WMMA)
//   -> exp -> [S1|S0] += E*[iv|1]  (8 chained WMMAs, C persists across loop)
// All LDS operand chunks are preloaded before each WMMA chain so operand
// latency overlaps the previous chain instead of serializing with it.
// ---------------------------------------------------------------------------
__global__ __launch_bounds__(BLOCK) void k_pass1(
    const float* __restrict__ grad, const float* __restrict__ sharp,
    const float* __restrict__ Wp, const float* __restrict__ bp,
    const float* __restrict__ Wk, const float* __restrict__ Wv,
    float* __restrict__ ws) {
  __shared__ float xb[NWAVE][32][9];    // x features, per element
  __shared__ float yb[NWAVE][32][17];   // [ik|iv] per element
  __shared__ float eb[NWAVE][16][33];   // E[m][elem]
  __shared__ float iqs[16][8];
  __shared__ float acc[144];

  const int tid  = threadIdx.x;
  const int wid  = tid >> 5;
  const int lane = tid & 31;
  const int half = lane >> 4;
  const int row  = lane & 15;

  if (tid < 128) iqs[tid >> 3][tid & 7] = ws[WS_IQ + tid];
  if (tid < 144) acc[tid] = 0.0f;
  __syncthreads();

  // Invariant WMMA operands. Layout: lane l, vgpr v -> K=4c+2*(l>>4)+v, N=l&15.
  v2f Bkv[2], Aiq[2];
  #pragma unroll
  for (int c = 0; c < 2; ++c) {
    #pragma unroll
    for (int v = 0; v < 2; ++v) {
      int k = 4 * c + 2 * half + v;
      Bkv[c][v] = (row < 8) ? Wk[row * 8 + k] : Wv[(row - 8) * 8 + k];
      Aiq[c][v] = iqs[row][k];
    }
  }
  float wp0[8], wp1[8], bpv[8];
  #pragma unroll
  for (int j = 0; j < 8; ++j) { wp0[j] = Wp[j * 2]; wp1[j] = Wp[j * 2 + 1]; bpv[j] = bp[j]; }

  v8f cS = {0.f, 0.f, 0.f, 0.f, 0.f, 0.f, 0.f, 0.f};  // rows m, cols: 0-7 = S1, 8 = S0

  for (int base = blockIdx.x * BLOCK + wid * 32; base < N_ELEM; base += GRID * BLOCK) {
    const int e = base + lane;
    float g = grad[e];
    float s = sharp[e];
    __builtin_prefetch(grad + e + GRID * BLOCK, 0, 1);
    __builtin_prefetch(sharp + e + GRID * BLOCK, 0, 1);

    // x = inp @ input_proj_W^T + b (K=2: cheaper per-lane than WMMA)
    #pragma unroll
    for (int j = 0; j < 8; ++j)
      xb[wid][lane][j] = fmaf(wp0[j], g, fmaf(wp1[j], s, bpv[j]));
    wave_lds_fence();

    // GEMM1: [ik|iv](32x16) = x(32x8) @ [Wk^T|Wv^T](8x16)
    v2f a1[4];                                   // all A chunks first
    #pragma unroll
    for (int t = 0; t < 2; ++t) {
      #pragma unroll
      for (int c = 0; c < 2; ++c) {
        a1[t * 2 + c][0] = xb[wid][t * 16 + row][4 * c + 2 * half + 0];
        a1[t * 2 + c][1] = xb[wid][t * 16 + row][4 * c + 2 * half + 1];
      }
    }
    v8f Ckv[2];
    #pragma unroll
    for (int t = 0; t < 2; ++t) {
      v8f cc = {0.f, 0.f, 0.f, 0.f, 0.f, 0.f, 0.f, 0.f};
      cc = wmma4(a1[t * 2 + 0], Bkv[0], cc);
      cc = wmma4(a1[t * 2 + 1], Bkv[1], cc);
      Ckv[t] = cc;
    }
    #pragma unroll
    for (int t = 0; t < 2; ++t) {
      #pragma unroll
      for (int v = 0; v < 8; ++v)
        yb[wid][t * 16 + v + 8 * half][row] = Ckv[t][v];
    }
    wave_lds_fence();

    // GEMM2: E(16x32) = iq(16x8) @ ik^T(8x32), then e = exp(scale*E)
    v2f b2[4];                                   // all B chunks first
    #pragma unroll
    for (int t = 0; t < 2; ++t) {
      #pragma unroll
      for (int c = 0; c < 2; ++c) {
        b2[t * 2 + c][0] = yb[wid][t * 16 + row][4 * c + 2 * half + 0];
        b2[t * 2 + c][1] = yb[wid][t * 16 + row][4 * c + 2 * half + 1];
      }
    }
    #pragma unroll
    for (int t = 0; t < 2; ++t) {
      v8f cc = {0.f, 0.f, 0.f, 0.f, 0.f, 0.f, 0.f, 0.f};
      cc = wmma4(Aiq[0], b2[t * 2 + 0], cc);
      cc = wmma4(Aiq[1], b2[t * 2 + 1], cc);
      #pragma unroll
      for (int v = 0; v < 8; ++v)
        eb[wid][v + 8 * half][t * 16 + row] = __expf(SCALE_D * cc[v]);
    }
    wave_lds_fence();

    // GEMM3: cS(16x16) += E(16x32) @ [iv|1](32x9..16pad), K=32 -> 8 chunks
    v2f a3[8], b3[8];                            // all operands first
    #pragma unroll
    for (int c = 0; c < 8; ++c) {
      int ke = 4 * c + 2 * half;
      a3[c][0] = eb[wid][row][ke + 0];
      a3[c][1] = eb[wid][row][ke + 1];
      #pragma unroll
      for (int v = 0; v < 2; ++v) {
        float bv;
        if (row < 8)       bv = yb[wid][ke + v][8 + row];   // iv component
        else if (row == 8) bv = 1.0f;                        // ones column -> S0
        else               bv = 0.0f;
        b3[c][v] = bv;
      }
    }
    #pragma unroll
    for (int c = 0; c < 8; ++c) cS = wmma4(a3[c], b3[c], cS);
    wave_lds_fence();
  }

  // Reduce cS: lane holds rows m=v+8*half at col n=row; cols 0..8 are live.
  if (row <= 8) {
    #pragma unroll
    for (int v = 0; v < 8; ++v)
      atomicAdd(&acc[(v + 8 * half) * 9 + row], cS[v]);
  }
  __syncthreads();
  if (tid < 144) {
    int m = tid / 9, n = tid % 9;
    float val = acc[tid];
    if (n < 8) atomicAdd(&ws[WS_S1 + m * 8 + n], val);
    else       atomicAdd(&ws[WS_S0 + m], val);
  }
}

// ---------------------------------------------------------------------------
// Pass 2: h_new/rq GEMM -> tanh -> read-attn scores GEMM -> softmax ->
// context GEMM -> PEER query GEMM -> product-key score GEMMs -> argmax ->
// expert gather + tiny MLP -> smart_grad.
// ---------------------------------------------------------------------------
__global__ __launch_bounds__(BLOCK) void k_pass2(
    const float* __restrict__ grad, const float* __restrict__ sharp,
    const float* __restrict__ hstate,
    const float* __restrict__ Wh, const float* __restrict__ Wx, const float* __restrict__ bx,
    const float* __restrict__ pq, const float* __restrict__ pkA, const float* __restrict__ pkB,
    const float* __restrict__ eW1, const float* __restrict__ eb1,
    const float* __restrict__ eW2, const float* __restrict__ eb2,
    const float* __restrict__ ws, float* __restrict__ out) {
  __shared__ float xb[NWAVE][32][21];   // A staging (K up to 20)
  __shared__ float yb[NWAVE][32][17];   // GEMM outputs per element
  __shared__ float ab[NWAVE][32][17];   // read-attention weights
  __shared__ float sb[NWAVE][32][33];   // 16 scores / 32 key scores per element
  __shared__ float iup[16][8];

  const int tid  = threadIdx.x;
  const int wid  = tid >> 5;
  const int lane = tid & 31;
  const int half = lane >> 4;
  const int row  = lane & 15;

  if (tid < 128) {
    int m = tid >> 3;
    iup[m][tid & 7] = ws[WS_S1 + tid] / ws[WS_S0 + m];
  }
  __syncthreads();

  // Invariant B operands
  v2f Bhn[3], Bis[2], Bic[4], Bq[5], Bka[2], Bkb[2];
  #pragma unroll
  for (int c = 0; c < 3; ++c) {                      // [h_new_pre | rq]: K = 12
    #pragma unroll
    for (int v = 0; v < 2; ++v) {
      int k = 4 * c + 2 * half + v;
      float bv = 0.0f;
      if (k < 8)        bv = (row < 8) ? Wh[row * 8 + k] : 0.0f;
      else if (k == 8)  bv = (row < 8) ? Wx[row * 2 + 0] : ws[WS_WRP + (row - 8) * 2 + 0];
      else if (k == 9)  bv = (row < 8) ? Wx[row * 2 + 1] : ws[WS_WRP + (row - 8) * 2 + 1];
      else if (k == 10) bv = (row < 8) ? bx[row]         : ws[WS_BRP + (row - 8)];
      Bhn[c][v] = bv;
    }
  }
  #pragma unroll
  for (int c = 0; c < 2; ++c) {                      // scores: rq @ I_up^T
    #pragma unroll
    for (int v = 0; v < 2; ++v) Bis[c][v] = iup[row][4 * c + 2 * half + v];
  }
  #pragma unroll
  for (int c = 0; c < 4; ++c) {                      // context: attn @ I_up
    #pragma unroll
    for (int v = 0; v < 2; ++v) {
      int k = 4 * c + 2 * half + v;
      Bic[c][v] = (row < 8) ? iup[k][row] : 0.0f;
    }
  }
  #pragma unroll
  for (int c = 0; c < 5; ++c) {                      // query: peer @ pq^T, K = 20 (18 live)
    #pragma unroll
    for (int v = 0; v < 2; ++v) {
      int k = 4 * c + 2 * half + v;
      Bq[c][v] = (k < 18 && row < 8) ? pq[row * 18 + k] : 0.0f;
    }
  }
  #pragma unroll
  for (int kb = 0; kb < 2; ++kb) {                   // product keys, K = 4
    #pragma unroll
    for (int v = 0; v < 2; ++v) {
      int k = 2 * half + v;
      Bka[kb][v] = pkA[(16 * kb + row) * 4 + k];
      Bkb[kb][v] = pkB[(16 * kb + row) * 4 + k];
    }
  }

  for (int base = blockIdx.x * BLOCK + wid * 32; base < N_ELEM; base += GRID * BLOCK) {
    const int e = base + lane;
    float g = grad[e];
    float s = sharp[e];
    float4 h0 = ((const float4*)hstate)[e * 2 + 0];
    float4 h1 = ((const float4*)hstate)[e * 2 + 1];
    __builtin_prefetch(hstate + (e + GRID * BLOCK) * 8, 0, 1);

    // Stage A = [h(8), g, s, 1, 0]
    xb[wid][lane][0] = h0.x; xb[wid][lane][1] = h0.y; xb[wid][lane][2] = h0.z; xb[wid][lane][3] = h0.w;
    xb[wid][lane][4] = h1.x; xb[wid][lane][5] = h1.y; xb[wid][lane][6] = h1.z; xb[wid][lane][7] = h1.w;
    xb[wid][lane][8] = g; xb[wid][lane][9] = s; xb[wid][lane][10] = 1.0f; xb[wid][lane][11] = 0.0f;
    wave_lds_fence();

    // GEMM: cols 0-7 = h_new pre-act, cols 8-15 = rq
    v2f ah[6];
    #pragma unroll
    for (int t = 0; t < 2; ++t) {
      #pragma unroll
      for (int c = 0; c < 3; ++c) {
        ah[t * 3 + c][0] = xb[wid][t * 16 + row][4 * c + 2 * half + 0];
        ah[t * 3 + c][1] = xb[wid][t * 16 + row][4 * c + 2 * half + 1];
      }
    }
    #pragma unroll
    for (int t = 0; t < 2; ++t) {
      v8f cc = {0.f, 0.f, 0.f, 0.f, 0.f, 0.f, 0.f, 0.f};
      cc = wmma4(ah[t * 3 + 0], Bhn[0], cc);
      cc = wmma4(ah[t * 3 + 1], Bhn[1], cc);
      cc = wmma4(ah[t * 3 + 2], Bhn[2], cc);
      #pragma unroll
      for (int v = 0; v < 8; ++v) yb[wid][t * 16 + v + 8 * half][row] = cc[v];
    }
    wave_lds_fence();

    // h_new = tanh(pre); emit h_new output
    float hn[8];
    #pragma unroll
    for (int j = 0; j < 8; ++j) hn[j] = tanhf(yb[wid][lane][j]);
    float4 o0 = {hn[0], hn[1], hn[2], hn[3]};
    float4 o1 = {hn[4], hn[5], hn[6], hn[7]};
    ((float4*)(out + N_ELEM))[e * 2 + 0] = o0;
    ((float4*)(out + N_ELEM))[e * 2 + 1] = o1;

    // GEMM: scores(32x16) = rq(32x8) @ I_up^T(8x16)   (A read from yb cols 8-15)
    v2f as[4];
    #pragma unroll
    for (int t = 0; t < 2; ++t) {
      #pragma unroll
      for (int c = 0; c < 2; ++c) {
        as[t * 2 + c][0] = yb[wid][t * 16 + row][8 + 4 * c + 2 * half + 0];
        as[t * 2 + c][1] = yb[wid][t * 16 + row][8 + 4 * c + 2 * half + 1];
      }
    }
    #pragma unroll
    for (int t = 0; t < 2; ++t) {
      v8f cc = {0.f, 0.f, 0.f, 0.f, 0.f, 0.f, 0.f, 0.f};
      cc = wmma4(as[t * 2 + 0], Bis[0], cc);
      cc = wmma4(as[t * 2 + 1], Bis[1], cc);
      #pragma unroll
      for (int v = 0; v < 8; ++v) sb[wid][t * 16 + v + 8 * half][row] = cc[v];
    }
    wave_lds_fence();

    // per-lane softmax over 16 inducing rows
    {
      float at[16], mx = -1e30f;
      #pragma unroll
      for (int m = 0; m < 16; ++m) { at[m] = SCALE_D * sb[wid][lane][m]; mx = fmaxf(mx, at[m]); }
      float ssum = 0.0f;
      #pragma unroll
      for (int m = 0; m < 16; ++m) { at[m] = __expf(at[m] - mx); ssum += at[m]; }
      float inv = 1.0f / ssum;
      #pragma unroll
      for (int m = 0; m < 16; ++m) ab[wid][lane][m] = at[m] * inv;
    }
    wave_lds_fence();

    // GEMM: context(32x8) = attn(32x16) @ I_up(16x8)
    v2f ac[8];
    #pragma unroll
    for (int t = 0; t < 2; ++t) {
      #pragma unroll
      for (int c = 0; c < 4; ++c) {
        ac[t * 4 + c][0] = ab[wid][t * 16 + row][4 * c + 2 * half + 0];
        ac[t * 4 + c][1] = ab[wid][t * 16 + row][4 * c + 2 * half + 1];
      }
    }
    #pragma unroll
    for (int t = 0; t < 2; ++t) {
      v8f cc = {0.f, 0.f, 0.f, 0.f, 0.f, 0.f, 0.f, 0.f};
      #pragma unroll
      for (int c = 0; c < 4; ++c) cc = wmma4(ac[t * 4 + c], Bic[c], cc);
      #pragma unroll
      for (int v = 0; v < 8; ++v) yb[wid][t * 16 + v + 8 * half][row] = cc[v];
    }
    wave_lds_fence();
    float cx[8];
    #pragma unroll
    for (int j = 0; j < 8; ++j) cx[j] = yb[wid][lane][j];

    // Stage A = peer_input = [h_new(8), context(8), g, s, 0, 0]
    #pragma unroll
    for (int j = 0; j < 8; ++j) { xb[wid][lane][j] = hn[j]; xb[wid][lane][8 + j] = cx[j]; }
    xb[wid][lane][16] = g; xb[wid][lane][17] = s; xb[wid][lane][18] = 0.0f; xb[wid][lane][19] = 0.0f;
    wave_lds_fence();

    // GEMM: query(32x8) = peer(32x18pad20) @ pq^T
    v2f aq[10];
    #pragma unroll
    for (int t = 0; t < 2; ++t) {
      #pragma unroll
      for (int c = 0; c < 5; ++c) {
        aq[t * 5 + c][0] = xb[wid][t * 16 + row][4 * c + 2 * half + 0];
        aq[t * 5 + c][1] = xb[wid][t * 16 + row][4 * c + 2 * half + 1];
      }
    }
    #pragma unroll
    for (int t = 0; t < 2; ++t) {
      v8f cc = {0.f, 0.f, 0.f, 0.f, 0.f, 0.f, 0.f, 0.f};
      #pragma unroll
      for (int c = 0; c < 5; ++c) cc = wmma4(aq[t * 5 + c], Bq[c], cc);
      #pragma unroll
      for (int v = 0; v < 8; ++v) yb[wid][t * 16 + v + 8 * half][row] = cc[v];
    }
    wave_lds_fence();

    // GEMM: scores_a(32x32) = q_a(32x4) @ pkA^T ; argmax
    v2f aka[2], akb[2];
    #pragma unroll
    for (int t = 0; t < 2; ++t) {
      aka[t][0] = yb[wid][t * 16 + row][2 * half + 0];
      aka[t][1] = yb[wid][t * 16 + row][2 * half + 1];
      akb[t][0] = yb[wid][t * 16 + row][4 + 2 * half + 0];
      akb[t][1] = yb[wid][t * 16 + row][4 + 2 * half + 1];
    }
    #pragma unroll
    for (int t = 0; t < 2; ++t) {
      #pragma unroll
      for (int kb = 0; kb < 2; ++kb) {
        v8f cc = {0.f, 0.f, 0.f, 0.f, 0.f, 0.f, 0.f, 0.f};
        cc = wmma4(aka[t], Bka[kb], cc);
        #pragma unroll
        for (int v = 0; v < 8; ++v) sb[wid][t * 16 + v + 8 * half][16 * kb + row] = cc[v];
      }
    }
    wave_lds_fence();
    int ia = 0;
    {
      float best = -1e30f;
      #pragma unroll
      for (int j = 0; j < 32; ++j) { float vv = sb[wid][lane][j]; if (vv > best) { best = vv; ia = j; } }
    }
    wave_lds_fence();

    // GEMM: scores_b = q_b(32x4) @ pkB^T ; argmax
    #pragma unroll
    for (int t = 0; t < 2; ++t) {
      #pragma unroll
      for (int kb = 0; kb < 2; ++kb) {
        v8f cc = {0.f, 0.f, 0.f, 0.f, 0.f, 0.f, 0.f, 0.f};
        cc = wmma4(akb[t], Bkb[kb], cc);
        #pragma unroll
        for (int v = 0; v < 8; ++v) sb[wid][t * 16 + v + 8 * half][16 * kb + row] = cc[v];
      }
    }
    wave_lds_fence();
    int ib = 0;
    {
      float best = -1e30f;
      #pragma unroll
      for (int j = 0; j < 32; ++j) { float vv = sb[wid][lane][j]; if (vv > best) { best = vv; ib = j; } }
    }
    wave_lds_fence();

    // Expert gather + tiny MLP (tables are 52KB -> L2 resident)
    int idx = ia * 32 + ib;
    float4 w1 = ((const float4*)eW1)[idx];
    float4 b1 = ((const float4*)eb1)[idx];
    float4 w2 = ((const float4*)eW2)[idx];
    float b2v = eb2[idx];
    float z0 = fmaxf(0.0f, fmaf(w1.x, g, b1.x));
    float z1 = fmaxf(0.0f, fmaf(w1.y, g, b1.y));
    float z2 = fmaxf(0.0f, fmaf(w1.z, g, b1.z));
    float z3 = fmaxf(0.0f, fmaf(w1.w, g, b1.w));
    float o = fmaf(w2.x, z0, fmaf(w2.y, z1, fmaf(w2.z, z2, w2.w * z3))) + b2v;
    out[e] = fmaf(RESCALE, o, g);
  }
}

extern "C" void kernel_launch(void* const* d_in, const int* in_sizes, int n_in,
                              void* d_out, int out_size, void* d_ws, size_t ws_size,
                              hipStream_t stream) {
  const float* grad   = (const float*)d_in[0];
  const float* sharp  = (const float*)d_in[1];
  const float* hstate = (const float*)d_in[2];
  const float* ip     = (const float*)d_in[3];
  const float* Wp     = (const float*)d_in[4];
  const float* bp     = (const float*)d_in[5];
  const float* Wq     = (const float*)d_in[6];
  const float* Wk     = (const float*)d_in[7];
  const float* Wv     = (const float*)d_in[8];
  const float* Wr     = (const float*)d_in[9];
  const float* Wh     = (const float*)d_in[10];
  const float* Wx     = (const float*)d_in[11];
  const float* bx     = (const float*)d_in[12];
  const float* pq     = (const float*)d_in[13];
  const float* pkA    = (const float*)d_in[14];
  const float* pkB    = (const float*)d_in[15];
  const float* eW1    = (const float*)d_in[16];
  const float* eb1    = (const float*)d_in[17];
  const float* eW2    = (const float*)d_in[18];
  const float* eb2    = (const float*)d_in[19];
  float* out = (float*)d_out;
  float* ws  = (float*)d_ws;

  k_setup<<<1, 256, 0, stream>>>(ip, Wq, Wr, Wp, bp, ws);
  k_pass1<<<GRID, BLOCK, 0, stream>>>(grad, sharp, Wp, bp, Wk, Wv, ws);
  k_pass2<<<GRID, BLOCK, 0, stream>>>(grad, sharp, hstate, Wh, Wx, bx, pq, pkA, pkB,
                                      eW1, eb1, eW2, eb2, ws, out);
}